// BarrierNet_84868553769084
// MI455X (gfx1250) — compile-verified
//
#include <hip/hip_runtime.h>
#include <cmath>

typedef float v2f __attribute__((ext_vector_type(2)));
typedef float v8f __attribute__((ext_vector_type(8)));

struct Smem {
  float xs[256][8];     // raw x, 256 samples
  float W1s[256][8];    // layer-1 weights
  float b1s[256];
  v2f   Wp[128][129];   // current branch W2 in K-pair layout, padded pitch
  float bias2[128];     // current branch b2
  float WoA[128];       // current branch Wout row 0
  float WoB[128];       // current branch Wout row 1
  float bo0, bo1;       // current branch bout
  float res1[256][2];   // x31
  float res2[256][2];   // x32
};

__device__ __forceinline__ void stage_branch(Smem& sm, const float* __restrict__ W2,
                                             const float* __restrict__ b2,
                                             const float* __restrict__ Wo,
                                             const float* __restrict__ bo, int tid) {
  // W2 is [128][256] row-major; float2 j of row n = (W2[n][2j], W2[n][2j+1]) = Wp[j][n]
  const v2f* wg = (const v2f*)W2;
  #pragma unroll 4
  for (int idx = tid; idx < 128 * 128; idx += 256) {
    int n  = idx >> 7;
    int k2 = idx & 127;
    sm.Wp[k2][n] = wg[idx];           // coalesced global, conflict-free LDS (pitch 129)
  }
  if (tid < 128) {
    sm.bias2[tid] = b2[tid];
    sm.WoA[tid]   = Wo[tid];
  } else {
    sm.WoB[tid - 128] = Wo[tid];
  }
  if (tid == 0) { sm.bo0 = bo[0]; sm.bo1 = bo[1]; }
}

template <bool SIG>
__device__ __forceinline__ void branch_pass(Smem& sm, float (*res)[2]) {
  const int tid  = threadIdx.x;
  const int lane = tid & 31;
  const int w    = tid >> 5;
  const int lm   = lane & 15;     // M (A) or N (B/C) index within tile
  const int h    = lane >> 4;     // lane-half selects K pair / M+8
  const int wb   = w * 32;        // this wave's local sample base (2 M-tiles)

  // per-lane copy of the 8 input features for the two tile samples
  float xa[8], xb[8];
  #pragma unroll
  for (int f = 0; f < 8; ++f) {
    xa[f] = sm.xs[wb + lm][f];
    xb[f] = sm.xs[wb + 16 + lm][f];
  }

  v8f C0[8], C1[8];
  #pragma unroll
  for (int nt = 0; nt < 8; ++nt) {
    #pragma unroll
    for (int q = 0; q < 8; ++q) { C0[nt][q] = 0.0f; C1[nt][q] = 0.0f; }
  }

  // K loop over the 256 hidden dims, 4 per WMMA step
  for (int k4 = 0; k4 < 64; ++k4) {
    const int kA = k4 * 4 + 2 * h;      // this lane-half's first K
    // layer-1 on the fly: hid[m][k] = relu(b1[k] + sum_f x[m][f]*W1[k][f])
    float h0a = sm.b1s[kA], h1a = sm.b1s[kA + 1];
    float h0b = h0a,        h1b = h1a;
    #pragma unroll
    for (int f = 0; f < 8; ++f) {
      float w0 = sm.W1s[kA][f];
      float w1 = sm.W1s[kA + 1][f];
      h0a = fmaf(xa[f], w0, h0a);
      h1a = fmaf(xa[f], w1, h1a);
      h0b = fmaf(xb[f], w0, h0b);
      h1b = fmaf(xb[f], w1, h1b);
    }
    v2f A0, A1;                          // A frag: K = r + 2*half
    A0.x = fmaxf(h0a, 0.0f); A0.y = fmaxf(h1a, 0.0f);
    A1.x = fmaxf(h0b, 0.0f); A1.y = fmaxf(h1b, 0.0f);

    const int kp = k4 * 2 + h;           // K-pair index for B frag
    #pragma unroll
    for (int nt = 0; nt < 8; ++nt) {
      v2f Bv = sm.Wp[kp][nt * 16 + lm];  // B frag: K = r + 2*half, N = lane%16
      C0[nt] = __builtin_amdgcn_wmma_f32_16x16x4_f32(false, A0, false, Bv,
                                                     (short)0, C0[nt], false, false);
      C1[nt] = __builtin_amdgcn_wmma_f32_16x16x4_f32(false, A1, false, Bv,
                                                     (short)0, C1[nt], false, false);
    }
  }

  // epilogue: bias + relu, fused [128 -> 2] projection, cross-lane reduce
  float acc0a[8], acc1a[8], acc0b[8], acc1b[8];
  #pragma unroll
  for (int r = 0; r < 8; ++r) { acc0a[r] = acc1a[r] = acc0b[r] = acc1b[r] = 0.0f; }

  #pragma unroll
  for (int nt = 0; nt < 8; ++nt) {
    const int n = nt * 16 + lm;
    const float bb = sm.bias2[n];
    const float w0 = sm.WoA[n];
    const float w1 = sm.WoB[n];
    #pragma unroll
    for (int r = 0; r < 8; ++r) {
      float va = fmaxf(C0[nt][r] + bb, 0.0f);   // a[m][n], m = r + 8*h
      float vb = fmaxf(C1[nt][r] + bb, 0.0f);
      acc0a[r] = fmaf(va, w0, acc0a[r]);
      acc1a[r] = fmaf(va, w1, acc1a[r]);
      acc0b[r] = fmaf(vb, w0, acc0b[r]);
      acc1b[r] = fmaf(vb, w1, acc1b[r]);
    }
  }

  const float bo0 = sm.bo0, bo1 = sm.bo1;
  #pragma unroll
  for (int r = 0; r < 8; ++r) {
    float a0 = acc0a[r], a1 = acc1a[r], c0 = acc0b[r], c1 = acc1b[r];
    #pragma unroll
    for (int m = 1; m <= 8; m <<= 1) {   // reduce within 16-lane half
      a0 += __shfl_xor(a0, m, 32);
      a1 += __shfl_xor(a1, m, 32);
      c0 += __shfl_xor(c0, m, 32);
      c1 += __shfl_xor(c1, m, 32);
    }
    if (lm == r) {                        // one writer per (half, r)
      float o0 = a0 + bo0, o1 = a1 + bo1;       // tile-0 sample m = r + 8h
      float p0 = c0 + bo0, p1 = c1 + bo1;       // tile-1 sample
      if (SIG) {
        o0 = 4.0f / (1.0f + expf(-o0));
        o1 = 4.0f / (1.0f + expf(-o1));
        p0 = 4.0f / (1.0f + expf(-p0));
        p1 = 4.0f / (1.0f + expf(-p1));
      }
      const int s0i = wb + r + 8 * h;
      const int s1i = wb + 16 + r + 8 * h;
      res[s0i][0] = o0; res[s0i][1] = o1;
      res[s1i][0] = p0; res[s1i][1] = p1;
    }
  }
}

__global__ __launch_bounds__(256)
void barriernet_kernel(const float* __restrict__ x,
                       const float* __restrict__ mean, const float* __restrict__ stdv,
                       const float* __restrict__ W1,  const float* __restrict__ b1,
                       const float* __restrict__ W21, const float* __restrict__ b21,
                       const float* __restrict__ W31, const float* __restrict__ b31,
                       const float* __restrict__ W22, const float* __restrict__ b22,
                       const float* __restrict__ W32, const float* __restrict__ b32,
                       float* __restrict__ out) {
  __shared__ Smem sm;
  const int tid = threadIdx.x;
  const int base = blockIdx.x * 256;

  // stage x / W1 / b1 (one row of 8 floats per thread, two float4 each)
  {
    const float4* xg = (const float4*)(x + (size_t)base * 8);
    float4 v0 = xg[tid * 2 + 0], v1 = xg[tid * 2 + 1];
    *(float4*)&sm.xs[tid][0] = v0; *(float4*)&sm.xs[tid][4] = v1;
    const float4* wg = (const float4*)W1;
    float4 u0 = wg[tid * 2 + 0], u1 = wg[tid * 2 + 1];
    *(float4*)&sm.W1s[tid][0] = u0; *(float4*)&sm.W1s[tid][4] = u1;
    sm.b1s[tid] = b1[tid];
  }
  stage_branch(sm, W21, b21, W31, b31, tid);
  __syncthreads();
  branch_pass<false>(sm, sm.res1);            // x31
  __syncthreads();
  stage_branch(sm, W22, b22, W32, b32, tid);
  __syncthreads();
  branch_pass<true>(sm, sm.res2);             // x32 = 4*sigmoid(...)
  __syncthreads();

  // ---------- phase 2: physics + closed-form QP, one thread per sample ----------
  float x0[8];
  #pragma unroll
  for (int f = 0; f < 8; ++f) x0[f] = sm.xs[tid][f] * stdv[f] + mean[f];
  const float p0 = sm.res1[tid][0], p1 = sm.res1[tid][1];
  const float s0 = sm.res2[tid][0], s1 = sm.res2[tid][1];

  const float px = x0[0], py = x0[1], th = x0[2], v = x0[3];
  const float ox = x0[4], oy = x0[5], oth = x0[6], ov = x0[7];
  const float st = sinf(th), ct = cosf(th);
  const float vs = v * st, vc = v * ct;
  const float ss = s0 + s1, sp = s0 * s1;

  float Gx[9], Gy[9], hh[9];
  const float R  = (0.2f + 0.5f) + 0.1f;   // AGENT_RADIUS + obs_r + SAFETY (f32 order as ref)
  const float RR = R * R;
  #pragma unroll
  for (int k = 0; k < 8; ++k) {
    // match numpy: float64 trig, cast to f32
    double ang = (double)k * 0.7853981633974483;   // 2*pi/8
    float obx = (float)(10.0 * cos(ang));
    float oby = (float)(10.0 * sin(ang));
    float dx = px - obx, dy = py - oby;
    float bar  = dx * dx + dy * dy - RR;
    float bdot = 2.0f * dx * vc + 2.0f * dy * vs;
    float Lf2b = 2.0f * v * v;
    float u1 = -2.0f * dx * vs + 2.0f * dy * vc;
    float u2 =  2.0f * dx * ct + 2.0f * dy * st;
    Gx[k] = -u1; Gy[k] = -u2;
    hh[k] = Lf2b + ss * bdot + sp * bar;
  }
  {
    const float Ro2 = 0.25f;                 // (2*0.2+0.1)^2
    float so = sinf(oth), co = cosf(oth);
    float dxo = px - ox, dyo = py - oy;
    float bar_o  = dxo * dxo + dyo * dyo - Ro2;
    float bdot_o = 2.0f * dxo * (vc - ov * co) + 2.0f * dyo * (vs - ov * so);
    float Lf2b_o = 2.0f * (v * v + ov * ov + 2.0f * v * ov * cosf(th - oth));
    float u1o = -2.0f * dxo * vs + 2.0f * dyo * vc;
    float u2o =  2.0f * dxo * ct + 2.0f * dyo * st;
    Gx[8] = -u1o; Gy[8] = -u2o;
    hh[8] = Lf2b_o + ss * bdot_o + sp * bar_o;
  }
  float tol[9];
  #pragma unroll
  for (int i = 0; i < 9; ++i) tol[i] = hh[i] + 1e-6f * (1.0f + fabsf(hh[i]));

  auto feas = [&](float zx, float zy) -> bool {
    bool ok = true;
    #pragma unroll
    for (int i = 0; i < 9; ++i) ok = ok && (zx * Gx[i] + zy * Gy[i] <= tol[i]);
    return ok;
  };

  // candidate 0: unconstrained
  float bx = -p0, by = -p1;
  float bestObj = __builtin_inff();
  if (feas(bx, by)) bestObj = 0.5f * (bx * bx + by * by) + bx * p0 + by * p1;

  // single active constraint (order i = 0..8, strict < keeps first min like argmin)
  #pragma unroll
  for (int i = 0; i < 9; ++i) {
    float gg  = Gx[i] * Gx[i] + Gy[i] * Gy[i];
    float lam = (-(Gx[i] * p0 + Gy[i] * p1) - hh[i]) / (gg + 1e-12f);
    float zx = -p0 - lam * Gx[i];
    float zy = -p1 - lam * Gy[i];
    if (lam >= -1e-8f && feas(zx, zy)) {
      float obj = 0.5f * (zx * zx + zy * zy) + zx * p0 + zy * p1;
      if (obj < bestObj) { bestObj = obj; bx = zx; by = zy; }
    }
  }
  // pairs, triu row-major order (0,1),(0,2)...(7,8)
  #pragma unroll
  for (int i = 0; i < 8; ++i) {
    #pragma unroll
    for (int j = i + 1; j < 9; ++j) {
      float det = Gx[i] * Gy[j] - Gy[i] * Gx[j];
      bool det_ok = fabsf(det) > 1e-9f;
      float ds = det_ok ? det : 1.0f;
      float zx = (hh[i] * Gy[j] - hh[j] * Gy[i]) / ds;
      float zy = (Gx[i] * hh[j] - Gx[j] * hh[i]) / ds;
      float rx = -(zx + p0), ry = -(zy + p1);
      float li = (Gy[j] * rx - Gx[j] * ry) / ds;
      float lj = (Gx[i] * ry - Gy[i] * rx) / ds;
      if (det_ok && li >= -1e-8f && lj >= -1e-8f && feas(zx, zy)) {
        float obj = 0.5f * (zx * zx + zy * zy) + zx * p0 + zy * p1;
        if (obj < bestObj) { bestObj = obj; bx = zx; by = zy; }
      }
    }
  }

  const int s = base + tid;
  out[2 * s + 0] = bx;
  out[2 * s + 1] = by;
}

extern "C" void kernel_launch(void* const* d_in, const int* in_sizes, int n_in,
                              void* d_out, int out_size, void* d_ws, size_t ws_size,
                              hipStream_t stream) {
  const float* x    = (const float*)d_in[0];
  const float* mean = (const float*)d_in[1];
  const float* stdv = (const float*)d_in[2];
  const float* W1   = (const float*)d_in[3];
  const float* b1   = (const float*)d_in[4];
  const float* W21  = (const float*)d_in[5];
  const float* b21  = (const float*)d_in[6];
  const float* W31  = (const float*)d_in[7];
  const float* b31  = (const float*)d_in[8];
  const float* W22  = (const float*)d_in[9];
  const float* b22  = (const float*)d_in[10];
  const float* W32  = (const float*)d_in[11];
  const float* b32  = (const float*)d_in[12];
  float* out = (float*)d_out;

  int B = in_sizes[0] / 8;          // 65536
  int grid = B / 256;               // 256 samples per block
  hipLaunchKernelGGL(barriernet_kernel, dim3(grid), dim3(256), 0, stream,
                     x, mean, stdv, W1, b1, W21, b21, W31, b31, W22, b22, W32, b32, out);
}